// BeliefModuleOld_51685636440379
// MI455X (gfx1250) — compile-verified
//
#include <hip/hip_runtime.h>
#include <cstdint>

// BeliefModule gather/select: B=1e6 elements, 240B treats + 20B vision in,
// 48B out per element. Pure bandwidth kernel (~308 MB => ~13us @ 23.3 TB/s).
// CDNA5 path: async global->LDS double-buffered staging (ASYNCcnt) so memory
// parallelism costs no VGPRs; branchless cndmask selection; b128 stores.

#define TPB 128                    // threads per block == batch rows per tile
#define ROW_BYTES 240              // 5*2*6 floats per batch element
#define TILE_BYTES (TPB * ROW_BYTES)  // 30720 B per tile
#define CHUNKS 15                  // 16B async chunks per thread per tile

__global__ __launch_bounds__(TPB) void belief_gather_kernel(
    const float* __restrict__ vt, const int* __restrict__ vis,
    float* __restrict__ out, int B)
{
  __shared__ __align__(16) unsigned char smem[2 * TILE_BYTES];  // 60 KB
  const int tid = (int)threadIdx.x;
  const int numTiles = (B + TPB - 1) / TPB;
  int tile = (int)blockIdx.x;
  if (tile >= numTiles) return;
  const int stride = (int)gridDim.x;

  // Low 32 bits of a generic LDS pointer are the wave-relative LDS byte offset.
  const uint32_t lbase = (uint32_t)(uintptr_t)&smem[0];
  const size_t totalBytes = (size_t)B * (size_t)ROW_BYTES;
  const unsigned char* gvt = (const unsigned char*)vt;

  // Issue one tile's worth of async global->LDS b128 transfers.
  // Every thread issues exactly CHUNKS ops => uniform per-wave ASYNCcnt.
  auto issue_tile = [&](int t_, int bufsel) {
    const unsigned char* gb = gvt + (size_t)t_ * (size_t)TILE_BYTES;  // uniform -> SGPR pair
    size_t rem = totalBytes - (size_t)t_ * (size_t)TILE_BYTES;
    uint32_t lim =
        (uint32_t)((rem >= (size_t)TILE_BYTES ? (size_t)TILE_BYTES : rem)) - 16u;
    uint32_t ldsa = lbase + (uint32_t)bufsel * (uint32_t)TILE_BYTES + (uint32_t)tid * 16u;
    uint32_t goff = (uint32_t)tid * 16u;
#pragma unroll
    for (int i = 0; i < CHUNKS; ++i) {
      uint32_t off = (goff > lim) ? lim : goff;  // tail tile: clamp source (dup reads, dead rows)
      asm volatile("global_load_async_to_lds_b128 %0, %1, %2"
                   :: "v"(ldsa), "v"(off), "s"(gb)
                   : "memory");
      ldsa += (uint32_t)(TPB * 16);
      goff += (uint32_t)(TPB * 16);
    }
  };

  issue_tile(tile, 0);  // prologue fill of buffer 0
  int buf = 0;
  for (; tile < numTiles; tile += stride, buf ^= 1) {
    const int nxt = tile + stride;
    const bool havenext = (nxt < numTiles);
    if (havenext) issue_tile(nxt, buf ^ 1);  // prefetch next tile into other buffer

    // Direct (non-async) vision loads for the CURRENT tile; latency hides
    // under the asynccnt wait + barrier below.
    const long long b = (long long)tile * TPB + tid;
    int v0 = 0, v1 = 0, v2 = 0, v3 = 0, v4 = 0;
    if (b < (long long)B) {
      const int* vp = vis + (size_t)b * 5u;
      v0 = vp[0]; v1 = vp[1]; v2 = vp[2]; v3 = vp[3]; v4 = vp[4];
    }

    // Async loads complete in order: waiting to <=15 retires the current
    // buffer's batch while the just-issued next batch stays in flight.
    if (havenext) asm volatile("s_wait_asynccnt 15" ::: "memory");
    else          asm volatile("s_wait_asynccnt 0"  ::: "memory");
    __syncthreads();  // publish all waves' staged LDS data

    if (b < (long long)B) {
      const float* row =
          (const float*)(smem + (size_t)buf * TILE_BYTES + (size_t)tid * ROW_BYTES);
      float r[12];
#pragma unroll
      for (int d = 0; d < 12; ++d) r[d] = 0.0f;
      r[5] = 1.0f;   // default for tt=0: [0,0,0,0,0,1]
      r[11] = 1.0f;  // default for tt=1
      const int seen[5] = {v0, v1, v2, v3, v4};
#pragma unroll
      for (int t = 0; t < 5; ++t) {
        const float* a0 = row + t * 12;      // [b, t, tt=0, 0..5]
        const float* a1 = a0 + 6;            // [b, t, tt=1, 0..5]
        float m0 = fmaxf(fmaxf(fmaxf(a0[0], a0[1]), fmaxf(a0[2], a0[3])), a0[4]);
        float m1 = fmaxf(fmaxf(fmaxf(a1[0], a1[1]), fmaxf(a1[2], a1[3])), a1[4]);
        const bool s0 = (seen[t] != 0) && (m0 > 0.5f);
        const bool s1 = (seen[t] != 0) && (m1 > 0.5f);
#pragma unroll
        for (int d = 0; d < 6; ++d) {        // ascending t: last valid wins
          r[d]     = s0 ? a0[d] : r[d];
          r[6 + d] = s1 ? a1[d] : r[6 + d];
        }
      }
      float4* op = (float4*)(out + (size_t)b * 12u);  // 48B, 16B-aligned
      op[0] = make_float4(r[0], r[1], r[2], r[3]);
      op[1] = make_float4(r[4], r[5], r[6], r[7]);
      op[2] = make_float4(r[8], r[9], r[10], r[11]);
    }
    __syncthreads();  // all waves done reading `buf` before it is re-filled
  }
}

extern "C" void kernel_launch(void* const* d_in, const int* in_sizes, int n_in,
                              void* d_out, int out_size, void* d_ws, size_t ws_size,
                              hipStream_t stream) {
  (void)n_in; (void)out_size; (void)d_ws; (void)ws_size;
  const float* vt = (const float*)d_in[0];  // [B,5,2,6] f32
  const int* vis  = (const int*)d_in[1];    // [B,5] i32
  float* out      = (float*)d_out;          // [B,2,6] f32
  const int B = in_sizes[0] / 60;           // 5*2*6 floats per element
  const int numTiles = (B + TPB - 1) / TPB;
  int blocks = numTiles < 2048 ? numTiles : 2048;  // persistent, double-buffered
  if (blocks < 1) blocks = 1;
  hipLaunchKernelGGL(belief_gather_kernel, dim3(blocks), dim3(TPB), 0, stream,
                     vt, vis, out, B);
}